// GAT_3676492005491
// MI455X (gfx1250) — compile-verified
//
#include <hip/hip_runtime.h>
#include <hip/hip_bf16.h>

typedef float v2f __attribute__((ext_vector_type(2)));
typedef float v4f __attribute__((ext_vector_type(4)));
typedef float v8f __attribute__((ext_vector_type(8)));

#define NEG_SLOPE 0.2f
#define ENC_NEG_INF 0x007FFFFFu   // encf(-inf)

// order-preserving float <-> uint32 encoding so segment-max can use atomicMax(u32)
__device__ __forceinline__ unsigned encf(float x) {
    unsigned u = __float_as_uint(x);
    return (u & 0x80000000u) ? ~u : (u | 0x80000000u);
}
__device__ __forceinline__ float decf(unsigned u) {
    return __uint_as_float((u & 0x80000000u) ? (u & 0x7FFFFFFFu) : ~u);
}
__device__ __forceinline__ float lrelu(float x) { return x > 0.f ? x : NEG_SLOPE * x; }
__device__ __forceinline__ void atomAddF(float* p, float v) {
    __hip_atomic_fetch_add(p, v, __ATOMIC_RELAXED, __HIP_MEMORY_SCOPE_AGENT);
}
__device__ __forceinline__ void atomMaxU(unsigned* p, unsigned v) {
    __hip_atomic_fetch_max(p, v, __ATOMIC_RELAXED, __HIP_MEMORY_SCOPE_AGENT);
}

// ---------------- init: accumulators to 0, max buffers to encoded -inf ----------------
__global__ __launch_bounds__(256) void gat_init(unsigned* m1, unsigned* m2,
                                                float* den1, float* den2,
                                                float* acc1, float* acc2, int Nn) {
    long long i = (long long)blockIdx.x * blockDim.x + threadIdx.x;
    long long n4 = (long long)Nn * 4, n128 = (long long)Nn * 128, n16 = (long long)Nn * 16;
    if (i < n4) { m1[i] = ENC_NEG_INF; }
    i -= n4;
    if (i >= 0 && i < Nn) { m2[i] = ENC_NEG_INF; }
    i -= Nn;
    if (i >= 0 && i < n4) { den1[i] = 0.f; }
    i -= n4;
    if (i >= 0 && i < Nn) { den2[i] = 0.f; }
    i -= Nn;
    if (i >= 0 && i < n128) { acc1[i] = 0.f; }
    i -= n128;
    if (i >= 0 && i < n16) { acc2[i] = 0.f; }
}

// ---------------- GEMM1: xw1[N,128] = x[N,256] @ W1[256,128], fp32 WMMA ----------------
// block = 256 thr = 8 waves; block tile = 16 rows x 128 cols; wave tile = 16x16, K step 4.
__global__ __launch_bounds__(256) void gat_gemm1(const float* __restrict__ x,
                                                 const float* __restrict__ W1,
                                                 float* __restrict__ xw1, int Nn) {
    __shared__ float tile[16 * 260];          // pad 260 (16B-aligned rows, bank-stagger 4)
    const int m0 = blockIdx.x * 16;
    const int t = threadIdx.x;
    const bool full = (m0 + 16 <= Nn);
    if (full) {
        // vectorized b128 fill: 1024 float4 chunks, 64 chunks per row
        for (int i = t; i < 1024; i += 256) {
            int r = i >> 6, c4 = (i & 63) << 2;
            *(v4f*)&tile[r * 260 + c4] = *(const v4f*)&x[(long long)(m0 + r) * 256 + c4];
        }
    } else {
        for (int i = t; i < 16 * 256; i += 256) {
            int r = i >> 8, c = i & 255;
            int gr = m0 + r; if (gr >= Nn) gr = Nn - 1;
            tile[r * 260 + c] = x[(long long)gr * 256 + c];
        }
    }
    __syncthreads();

    const int lane = t & 31, wave = t >> 5;
    const int n0 = wave * 16;
    const int mrow = lane & 15;               // A: lane -> M row
    const int kg   = (lane >> 4) << 1;        // K sub-offset 0 or 2 by lane half
    const int ncol = lane & 15;               // B/C/D: lane -> N col
    v8f c = {};
#pragma unroll 8
    for (int k = 0; k < 256; k += 4) {
        v2f a, b;
        a.x = tile[mrow * 260 + k + kg];
        a.y = tile[mrow * 260 + k + kg + 1];
        b.x = W1[(k + kg) * 128 + n0 + ncol];         // W1 is L2-resident (128 KB)
        b.y = W1[(k + kg + 1) * 128 + n0 + ncol];
        c = __builtin_amdgcn_wmma_f32_16x16x4_f32(false, a, false, b, (short)0, c,
                                                  false, false);
    }
    const int rbase = (lane >> 4) * 8;        // C/D: VGPR r -> M = r (+8 for hi lanes)
    float* outp = xw1 + (long long)(m0 + rbase) * 128 + n0 + ncol;
    if (full) {
#pragma unroll
        for (int r = 0; r < 8; ++r) outp[r * 128] = c[r];
    } else {
#pragma unroll
        for (int r = 0; r < 8; ++r)
            if (m0 + rbase + r < Nn) outp[r * 128] = c[r];
    }
}

// ---------------- per-node attention logits, layer 1 ----------------
__global__ __launch_bounds__(256) void gat_logits1(const float* __restrict__ xw1,
                                                   const float* __restrict__ att_s,
                                                   const float* __restrict__ att_d,
                                                   float* __restrict__ as1,
                                                   float* __restrict__ ad1, int Nn) {
    int tid = blockIdx.x * blockDim.x + threadIdx.x;
    if (tid >= Nn * 4) return;
    int n = tid >> 2, h = tid & 3;
    const float* row = xw1 + (long long)n * 128 + h * 32;
    float ss = 0.f, sd = 0.f;
#pragma unroll 8
    for (int d = 0; d < 32; ++d) {
        float v = row[d];
        ss += v * att_s[h * 32 + d];
        sd += v * att_d[h * 32 + d];
    }
    as1[tid] = ss;
    ad1[tid] = sd;
}

// ---------------- edge pass: segment max (layer 1), 1 thread per (edge,head) ----------
__global__ __launch_bounds__(256) void gat_edge_max1(const int* __restrict__ ei,
                                                     const float* __restrict__ as1,
                                                     const float* __restrict__ ad1,
                                                     unsigned* __restrict__ m1,
                                                     int Ee, int Nn) {
    long long tid = (long long)blockIdx.x * blockDim.x + threadIdx.x;
    long long tot = (long long)(Ee + Nn) * 4;
    if (tid >= tot) return;
    int e = (int)(tid >> 2), h = (int)(tid & 3);
    int s, d;
    if (e < Ee) { s = ei[e]; d = ei[Ee + e]; } else { s = d = e - Ee; }
    float v = lrelu(as1[s * 4 + h] + ad1[d * 4 + h]);
    atomMaxU(&m1[d * 4 + h], encf(v));
}

// ------- edge pass: exp + denom + message scatter (layer 1), 1 wave per edge ----------
__global__ __launch_bounds__(256) void gat_edge_acc1(const int* __restrict__ ei,
                                                     const float* __restrict__ as1,
                                                     const float* __restrict__ ad1,
                                                     const unsigned* __restrict__ m1,
                                                     const float* __restrict__ xw1,
                                                     float* __restrict__ den1,
                                                     float* __restrict__ acc1,
                                                     int Ee, int Nn) {
    long long tid = (long long)blockIdx.x * blockDim.x + threadIdx.x;
    int lane = (int)(tid & 31);
    long long e = tid >> 5;
    if (e >= (long long)(Ee + Nn)) return;
    int s, d;
    if (e < Ee) { s = ei[e]; d = ei[Ee + (int)e]; } else { s = d = (int)e - Ee; }
    float exv = 0.f;
    if (lane < 4) {                      // lanes 0-3: per-head exp(e - max), denom add
        int h = lane;
        float v = lrelu(as1[s * 4 + h] + ad1[d * 4 + h]);
        exv = __expf(v - decf(m1[d * 4 + h]));
        atomAddF(&den1[d * 4 + h], exv);
    }
    const float* src = xw1 + (long long)s * 128;
    float* dst = acc1 + (long long)d * 128;
#pragma unroll
    for (int c = 0; c < 4; ++c) {        // head c covers features [32c, 32c+32)
        float ec = __shfl(exv, c, 32);
        atomAddF(&dst[c * 32 + lane], ec * src[c * 32 + lane]);
    }
}

// ------ finalize layer1: h = relu(acc/denom + b1), in place, float4-wide ------
__global__ __launch_bounds__(256) void gat_final1(float* __restrict__ acc1,
                                                  const float* __restrict__ den1,
                                                  const float* __restrict__ b1, int Nn) {
    long long tid = (long long)blockIdx.x * blockDim.x + threadIdx.x;   // one float4
    if (tid >= (long long)Nn * 32) return;
    int n = (int)(tid >> 5), f4 = (int)(tid & 31);     // f4*4 = feature base
    int h = f4 >> 3;                                    // 8 float4s per head
    float inv = 1.f / den1[n * 4 + h];
    v4f v = *(v4f*)&acc1[(long long)n * 128 + f4 * 4];
    v4f bb = *(const v4f*)&b1[f4 * 4];
#pragma unroll
    for (int j = 0; j < 4; ++j) {
        float t = v[j] * inv + bb[j];
        v[j] = t > 0.f ? t : 0.f;
    }
    *(v4f*)&acc1[(long long)n * 128 + f4 * 4] = v;
}

// ---------------- GEMM2: xw2[N,16] = h[N,128] @ W2[128,16], fp32 WMMA ----------------
__global__ __launch_bounds__(256) void gat_gemm2(const float* __restrict__ h,
                                                 const float* __restrict__ W2,
                                                 float* __restrict__ xw2, int Ntiles,
                                                 int Nn) {
    int t = threadIdx.x, lane = t & 31, wave = t >> 5;
    int tileId = blockIdx.x * 8 + wave;
    if (tileId >= Ntiles) return;            // wave-uniform exit, EXEC stays all-1s
    int m0 = tileId * 16;
    const int mrow = lane & 15, kg = (lane >> 4) << 1, ncol = lane & 15;
    const bool full = (m0 + 16 <= Nn);
    int gr = m0 + mrow; if (gr >= Nn) gr = Nn - 1;
    const float* arow = h + (long long)gr * 128 + kg;
    v8f c = {};
#pragma unroll 8
    for (int k = 0; k < 128; k += 4) {
        v2f a, b;
        a.x = arow[k];
        a.y = arow[k + 1];
        b.x = W2[(k + kg) * 16 + ncol];
        b.y = W2[(k + kg + 1) * 16 + ncol];
        c = __builtin_amdgcn_wmma_f32_16x16x4_f32(false, a, false, b, (short)0, c,
                                                  false, false);
    }
    const int rbase = (lane >> 4) * 8;
    float* outp = xw2 + (long long)(m0 + rbase) * 16 + ncol;
    if (full) {
#pragma unroll
        for (int r = 0; r < 8; ++r) outp[r * 16] = c[r];
    } else {
#pragma unroll
        for (int r = 0; r < 8; ++r)
            if (m0 + rbase + r < Nn) outp[r * 16] = c[r];
    }
}

// ---------------- per-node logits, layer 2 (1 head, 16 dims) ----------------
__global__ __launch_bounds__(256) void gat_logits2(const float* __restrict__ xw2,
                                                   const float* __restrict__ att_s,
                                                   const float* __restrict__ att_d,
                                                   float* __restrict__ as2,
                                                   float* __restrict__ ad2, int Nn) {
    int n = blockIdx.x * blockDim.x + threadIdx.x;
    if (n >= Nn) return;
    const float* row = xw2 + (long long)n * 16;
    float ss = 0.f, sd = 0.f;
#pragma unroll
    for (int d = 0; d < 16; ++d) { ss += row[d] * att_s[d]; sd += row[d] * att_d[d]; }
    as2[n] = ss;
    ad2[n] = sd;
}

__global__ __launch_bounds__(256) void gat_edge_max2(const int* __restrict__ ei,
                                                     const float* __restrict__ as2,
                                                     const float* __restrict__ ad2,
                                                     unsigned* __restrict__ m2,
                                                     int Ee, int Nn) {
    long long e = (long long)blockIdx.x * blockDim.x + threadIdx.x;
    if (e >= (long long)(Ee + Nn)) return;
    int s, d;
    if (e < Ee) { s = ei[e]; d = ei[Ee + (int)e]; } else { s = d = (int)e - Ee; }
    atomMaxU(&m2[d], encf(lrelu(as2[s] + ad2[d])));
}

// 16 lanes per edge
__global__ __launch_bounds__(256) void gat_edge_acc2(const int* __restrict__ ei,
                                                     const float* __restrict__ as2,
                                                     const float* __restrict__ ad2,
                                                     const unsigned* __restrict__ m2,
                                                     const float* __restrict__ xw2,
                                                     float* __restrict__ den2,
                                                     float* __restrict__ acc2,
                                                     int Ee, int Nn) {
    long long tid = (long long)blockIdx.x * blockDim.x + threadIdx.x;
    int lane = (int)(tid & 15);
    long long e = tid >> 4;
    if (e >= (long long)(Ee + Nn)) return;
    int s, d;
    if (e < Ee) { s = ei[e]; d = ei[Ee + (int)e]; } else { s = d = (int)e - Ee; }
    float v = lrelu(as2[s] + ad2[d]);
    float ex = __expf(v - decf(m2[d]));
    if (lane == 0) atomAddF(&den2[d], ex);
    atomAddF(&acc2[(long long)d * 16 + lane], ex * xw2[(long long)s * 16 + lane]);
}

// ---------------- finalize layer2 + log_softmax over 16 classes ----------------
__global__ __launch_bounds__(256) void gat_final2(const float* __restrict__ acc2,
                                                  const float* __restrict__ den2,
                                                  const float* __restrict__ b2,
                                                  float* __restrict__ out, int Nn) {
    int n = blockIdx.x * blockDim.x + threadIdx.x;
    if (n >= Nn) return;
    float inv = 1.f / den2[n];
    float v[16], m = -1e30f;
#pragma unroll
    for (int c = 0; c < 16; ++c) {
        v[c] = acc2[(long long)n * 16 + c] * inv + b2[c];
        m = v[c] > m ? v[c] : m;
    }
    float s = 0.f;
#pragma unroll
    for (int c = 0; c < 16; ++c) s += __expf(v[c] - m);
    float lse = m + __logf(s);
#pragma unroll
    for (int c = 0; c < 16; ++c) out[(long long)n * 16 + c] = v[c] - lse;
}

extern "C" void kernel_launch(void* const* d_in, const int* in_sizes, int n_in,
                              void* d_out, int out_size, void* d_ws, size_t ws_size,
                              hipStream_t stream) {
    const float* x    = (const float*)d_in[0];
    const int*   ei   = (const int*)d_in[1];
    const float* W1   = (const float*)d_in[2];
    const float* at_s1= (const float*)d_in[3];
    const float* at_d1= (const float*)d_in[4];
    const float* b1   = (const float*)d_in[5];
    const float* W2   = (const float*)d_in[6];
    const float* at_s2= (const float*)d_in[7];
    const float* at_d2= (const float*)d_in[8];
    const float* b2   = (const float*)d_in[9];
    float* out = (float*)d_out;

    const int Nn = in_sizes[0] / 256;      // 50000
    const int Ee = in_sizes[1] / 2;        // 1600000
    const long long ET = (long long)Ee + Nn;

    // workspace layout (floats), total ~62 MB
    float* ws = (float*)d_ws;
    float*    xw1  = ws;                               // N*128
    float*    as1  = xw1 + (long long)Nn * 128;        // N*4
    float*    ad1  = as1 + (long long)Nn * 4;          // N*4
    unsigned* m1   = (unsigned*)(ad1 + (long long)Nn * 4); // N*4
    float*    den1 = (float*)(m1 + (long long)Nn * 4); // N*4
    float*    acc1 = den1 + (long long)Nn * 4;         // N*128 (becomes h)
    float*    xw2  = acc1 + (long long)Nn * 128;       // N*16
    float*    as2  = xw2 + (long long)Nn * 16;         // N
    float*    ad2  = as2 + Nn;                         // N
    unsigned* m2   = (unsigned*)(ad2 + Nn);            // N
    float*    den2 = (float*)(m2 + Nn);                // N
    float*    acc2 = den2 + Nn;                        // N*16

    const int B = 256;
    auto blk = [](long long n, int b) { return (unsigned)((n + b - 1) / b); };

    long long initN = (long long)Nn * (4 + 1 + 4 + 1 + 128 + 16);
    gat_init<<<blk(initN, B), B, 0, stream>>>(m1, m2, den1, den2, acc1, acc2, Nn);

    gat_gemm1<<<blk((long long)Nn, 16), B, 0, stream>>>(x, W1, xw1, Nn);
    gat_logits1<<<blk((long long)Nn * 4, B), B, 0, stream>>>(xw1, at_s1, at_d1, as1, ad1, Nn);
    gat_edge_max1<<<blk(ET * 4, B), B, 0, stream>>>(ei, as1, ad1, m1, Ee, Nn);
    gat_edge_acc1<<<blk(ET * 32, B), B, 0, stream>>>(ei, as1, ad1, m1, xw1, den1, acc1, Ee, Nn);
    gat_final1<<<blk((long long)Nn * 32, B), B, 0, stream>>>(acc1, den1, b1, Nn);

    int ntiles = (Nn + 15) / 16;
    gat_gemm2<<<blk((long long)ntiles, 8), B, 0, stream>>>(acc1, W2, xw2, ntiles, Nn);
    gat_logits2<<<blk((long long)Nn, B), B, 0, stream>>>(xw2, at_s2, at_d2, as2, ad2, Nn);
    gat_edge_max2<<<blk(ET, B), B, 0, stream>>>(ei, as2, ad2, m2, Ee, Nn);
    gat_edge_acc2<<<blk(ET * 16, B), B, 0, stream>>>(ei, as2, ad2, m2, xw2, den2, acc2, Ee, Nn);
    gat_final2<<<blk((long long)Nn, B), B, 0, stream>>>(acc2, den2, b2, out, Nn);
}